// SSIM_37417755083665
// MI455X (gfx1250) — compile-verified
//
#include <hip/hip_runtime.h>

typedef float v2f __attribute__((ext_vector_type(2)));
typedef float v8f __attribute__((ext_vector_type(8)));

#define TILE      32
#define HALO      5
#define IN_DIM    48          // 42 needed, padded to 3 rowgroups of 16
#define IN_PITCH  49          // odd pitch -> conflict-free A-fragment loads
#define HC_ROWS   48
#define HC_PITCH  33          // odd pitch -> conflict-free Hc access
#define KSZ       11
#define C1C       1.0e-4f
#define C2C       9.0e-4f
#define EPSC      1.0e-8f

#define WMMA_F32_16x16x4(A, B, C) \
    __builtin_amdgcn_wmma_f32_16x16x4_f32(false, (A), false, (B), (short)0, (C), false, false)

__global__ __launch_bounds__(128) void ssim_main_kernel(
    const float* __restrict__ pred, const float* __restrict__ targ,
    const float* __restrict__ kern, float* __restrict__ partial)
{
    __shared__ float sP[IN_DIM * IN_PITCH];
    __shared__ float sT[IN_DIM * IN_PITCH];
    __shared__ float sHc[5][HC_ROWS * HC_PITCH];
    __shared__ float sW[16];
    __shared__ float sRed[4];

    const int tid  = threadIdx.x;
    const int lane = tid & 31;
    const int wave = tid >> 5;
    const int m    = lane & 15;       // row (A/D) or column (B) index
    const int kh   = (lane >> 4) & 1; // K-half / row-half select

    const int plane   = blockIdx.y;           // b*3 + c
    const int tileIdx = blockIdx.x;           // 16x16 tiles of 32x32
    const int gy0 = (tileIdx >> 4) * TILE;
    const int gx0 = (tileIdx & 15) * TILE;
    const size_t pOff = (size_t)plane * (512 * 512);

    // --- 1D gaussian weights from row 5 of the 2D kernel (separable) ---
    if (tid < 16) sW[tid] = (tid < KSZ) ? kern[5 * KSZ + tid] : 0.0f;
    __syncthreads();
    float wsum = 0.0f;
    #pragma unroll
    for (int j = 0; j < KSZ; ++j) wsum += sW[j];
    const float wnorm = 1.0f / wsum;

    // Banded weight fragments: wf[q] = w[(4q+2*kh+{0,1}) - m], zero outside band.
    // Serves as B-operand (horizontal) and A-operand (vertical) identically.
    v2f wf[7];
    #pragma unroll
    for (int q = 0; q < 7; ++q) {
        const int j0 = 4 * q + 2 * kh;
        const int i0 = j0 - m, i1 = j0 + 1 - m;
        float x0 = ((unsigned)i0 <= 10u) ? sW[i0 & 15] * wnorm : 0.0f;
        float x1 = ((unsigned)i1 <= 10u) ? sW[i1 & 15] * wnorm : 0.0f;
        wf[q].x = x0; wf[q].y = x1;
    }

    // --- load zero-padded 48x48 input tiles, map [-1,1] -> [0,1] ---
    for (int i = tid; i < IN_DIM * IN_DIM; i += 128) {
        const int r = i / IN_DIM, c = i % IN_DIM;
        const int gr = gy0 - HALO + r, gc = gx0 - HALO + c;
        const bool ok = ((unsigned)gr < 512u) && ((unsigned)gc < 512u);
        const size_t gi = pOff + (size_t)(ok ? gr : 0) * 512 + (ok ? gc : 0);
        const float pv = ok ? pred[gi] : -1.0f;   // maps to 0 after transform
        const float tv = ok ? targ[gi] : -1.0f;
        sP[r * IN_PITCH + c] = (pv + 1.0f) * 0.5f;
        sT[r * IN_PITCH + c] = (tv + 1.0f) * 0.5f;
    }
    __syncthreads();

    // --- horizontal pass: Hc(48x32, 5 maps) = InTile x W_band via WMMA ---
    // 6 blocks of 16x16, wave-uniform distribution over 4 waves.
    for (int blk = wave; blk < 6; blk += 4) {
        const int r0 = (blk >> 1) * 16;
        const int c0 = (blk & 1) * 16;
        v8f aP = {}, aT = {}, aPP = {}, aTT = {}, aPT = {};
        #pragma unroll
        for (int q = 0; q < 7; ++q) {
            const int col  = c0 + 4 * q + 2 * kh;
            const int base = (r0 + m) * IN_PITCH + col;
            v2f ap; ap.x = sP[base]; ap.y = sP[base + 1];
            v2f at; at.x = sT[base]; at.y = sT[base + 1];
            const v2f app = ap * ap, att = at * at, apt = ap * at;
            aP  = WMMA_F32_16x16x4(ap,  wf[q], aP);
            aT  = WMMA_F32_16x16x4(at,  wf[q], aT);
            aPP = WMMA_F32_16x16x4(app, wf[q], aPP);
            aTT = WMMA_F32_16x16x4(att, wf[q], aTT);
            aPT = WMMA_F32_16x16x4(apt, wf[q], aPT);
        }
        #pragma unroll
        for (int v = 0; v < 8; ++v) {
            const int row = r0 + v + 8 * kh;
            const int o   = row * HC_PITCH + c0 + m;
            sHc[0][o] = aP[v];  sHc[1][o] = aT[v];
            sHc[2][o] = aPP[v]; sHc[3][o] = aTT[v]; sHc[4][o] = aPT[v];
        }
    }
    __syncthreads();

    // --- vertical pass: one 16x16 output subtile per wave ---
    const int r0v = (wave >> 1) * 16;
    const int c0v = (wave & 1) * 16;
    v8f mX = {}, mY = {}, eXX = {}, eYY = {}, eXY = {};
    #pragma unroll
    for (int q = 0; q < 7; ++q) {
        const int row = r0v + 4 * q + 2 * kh;
        const int o0 = row * HC_PITCH + c0v + m;
        const int o1 = o0 + HC_PITCH;
        v2f b0; b0.x = sHc[0][o0]; b0.y = sHc[0][o1];
        v2f b1; b1.x = sHc[1][o0]; b1.y = sHc[1][o1];
        v2f b2; b2.x = sHc[2][o0]; b2.y = sHc[2][o1];
        v2f b3; b3.x = sHc[3][o0]; b3.y = sHc[3][o1];
        v2f b4; b4.x = sHc[4][o0]; b4.y = sHc[4][o1];
        mX  = WMMA_F32_16x16x4(wf[q], b0, mX);
        mY  = WMMA_F32_16x16x4(wf[q], b1, mY);
        eXX = WMMA_F32_16x16x4(wf[q], b2, eXX);
        eYY = WMMA_F32_16x16x4(wf[q], b3, eYY);
        eXY = WMMA_F32_16x16x4(wf[q], b4, eXY);
    }

    // --- SSIM formula per element + reduction ---
    float s = 0.0f;
    #pragma unroll
    for (int v = 0; v < 8; ++v) {
        const float mx = mX[v], my = mY[v];
        const float sx  = eXX[v] - mx * mx;
        const float sy  = eYY[v] - my * my;
        const float sxy = eXY[v] - mx * my;
        const float num = (2.0f * mx * my + C1C) * (2.0f * sxy + C2C);
        const float den = (mx * mx + my * my + C1C) * (sx + sy + C2C);
        s += num / (den + EPSC);
    }
    #pragma unroll
    for (int off = 16; off > 0; off >>= 1) s += __shfl_xor(s, off, 32);
    if (lane == 0) sRed[wave] = s;
    __syncthreads();
    if (tid == 0)
        partial[(size_t)plane * 256 + tileIdx] = sRed[0] + sRed[1] + sRed[2] + sRed[3];
}

// Deterministic per-image reduction: 768 partials (3 planes x 256 tiles) per image.
__global__ __launch_bounds__(256) void ssim_reduce_kernel(
    const float* __restrict__ partial, float* __restrict__ out)
{
    __shared__ float sm[256];
    const int b = blockIdx.x, t = threadIdx.x;
    const float* p = partial + (size_t)b * 768;
    sm[t] = p[t] + p[t + 256] + p[t + 512];
    __syncthreads();
    #pragma unroll
    for (int off = 128; off > 0; off >>= 1) {
        if (t < off) sm[t] += sm[t + off];
        __syncthreads();
    }
    if (t == 0) out[b] = sm[0] * (1.0f / 786432.0f);  // mean over C*H*W
}

extern "C" void kernel_launch(void* const* d_in, const int* in_sizes, int n_in,
                              void* d_out, int out_size, void* d_ws, size_t ws_size,
                              hipStream_t stream)
{
    (void)in_sizes; (void)n_in; (void)out_size; (void)ws_size;
    const float* pred = (const float*)d_in[0];
    const float* targ = (const float*)d_in[1];
    const float* kern = (const float*)d_in[2];
    float* out     = (float*)d_out;
    float* partial = (float*)d_ws;   // 96 planes * 256 tiles * 4B = 98304 B

    dim3 grid(256, 96);              // 16x16 tiles per 512x512 plane, 96 planes
    ssim_main_kernel<<<grid, 128, 0, stream>>>(pred, targ, kern, partial);
    ssim_reduce_kernel<<<32, 256, 0, stream>>>(partial, out);
}